// HGTLayer_5514738008388
// MI455X (gfx1250) — compile-verified
//
#include <hip/hip_runtime.h>
#include <hip/hip_bf16.h>
#include <math.h>

// ---------------------------------------------------------------------------
// HGT layer for MI455X (gfx1250, wave32).
//  - Dense GEMMs on V_WMMA_F32_16X16X4_F32 (fp32 in/out: layer is HBM-bound
//    at 23.3 TB/s, so fp32 matches the reference at no bandwidth cost).
//  - Per-head relation transforms folded into projection weights
//    (block-diagonal fold) -> saves a 184MB intermediate pass.
//  - Weights pre-swizzled into per-lane consumption order so the GEMM inner
//    loop does 2x global_load_b128 (B) + 1x global_load_b64 (A) per 4 WMMAs.
//  - Edge softmax: wave-per-edge (32 lanes == one head), keyed atomicMax
//    segment-max, atomicAdd denominators / scatter (L2-side float atomics).
// ---------------------------------------------------------------------------

#define DIM   256
#define HEADS 8
#define DK    32

typedef __attribute__((ext_vector_type(2))) float v2f;
typedef __attribute__((ext_vector_type(4))) float v4f;
typedef __attribute__((ext_vector_type(8))) float v8f;

// ---------------------------------------------------------------------------
// Fold per-head relation matrix R [8][32][32] into weight W [256][256] and
// bias b [256]:  W'[k][n] = sum_d W[k][h*32+d] * R[h][d][e]  (h=n/32, e=n%32)
// ---------------------------------------------------------------------------
__global__ __launch_bounds__(256) void fuse_weight_kernel(
    const float* __restrict__ W, const float* __restrict__ b,
    const float* __restrict__ R, float* __restrict__ Wp, float* __restrict__ bp)
{
    int idx = blockIdx.x * blockDim.x + threadIdx.x;
    if (idx >= 257 * 256) return;
    int r = idx >> 8;        // 0..255 weight rows, 256 = bias row
    int n = idx & 255;
    int h = n >> 5, e = n & 31;
    const float* Rh  = R + h * (DK * DK);
    const float* src = (r < 256) ? (W + (size_t)r * DIM + h * DK) : (b + h * DK);
    float acc = 0.f;
#pragma unroll 8
    for (int d = 0; d < DK; ++d) acc += src[d] * Rh[d * DK + e];
    if (r < 256) Wp[(size_t)r * DIM + n] = acc;
    else         bp[n] = acc;
}

// ---------------------------------------------------------------------------
// Swizzle B[256][256] into per-lane WMMA consumption order:
//   Bsw[((ng*64 + ks)*32 + lane)*8 + {b0.x,b0.y,b1.x,b1.y,b2.x,b2.y,b3.x,b3.y}]
// where b_t = (B[kr][c], B[kr+1][c]), kr = 4*ks + 2*(lane>=16),
//       c = ng*64 + (lane&15) + 16*t.
// A wave then reads its 8 B operand floats per k-step as two b128 loads of a
// fully coalesced 1KB stream.
// ---------------------------------------------------------------------------
__global__ __launch_bounds__(256) void swizzle_B_kernel(
    const float* __restrict__ B, float* __restrict__ Bsw)
{
    int i = blockIdx.x * blockDim.x + threadIdx.x;   // 4 ng * 64 ks * 32 lanes
    if (i >= 4 * 64 * 32) return;
    const int lane = i & 31;
    const int ks   = (i >> 5) & 63;
    const int ng   = i >> 11;
    const int kr   = (ks << 2) + ((lane >> 4) << 1);
    const int c0   = (ng << 6) + (lane & 15);
    float* o = Bsw + (size_t)i * 8;
#pragma unroll
    for (int t = 0; t < 4; ++t) {
        const int c = c0 + (t << 4);
        o[2 * t + 0] = B[(size_t)kr * DIM + c];
        o[2 * t + 1] = B[(size_t)(kr + 1) * DIM + c];
    }
}

// ---------------------------------------------------------------------------
// C[M x 256] = (ascale*A)[M x 256] @ B[256 x 256] + bias
// SKIP epilogue: C = C*sigmoid(skip[skipIdx]) + resid*(1-sigmoid(...))
// One wave -> 16x64 output stripe (4 f32 accumulators), M multiple of 16.
// WMMA f32 16x16x4 layouts per ISA: A = 2 VGPRs (lanes 0-15: K=0,1;
// lanes 16-31: K=2,3); C/D: VGPR j = rows j / j+8.
// ---------------------------------------------------------------------------
template <bool SCALE, bool SKIP>
__global__ __launch_bounds__(256) void gemm256_wmma_kernel(
    const float* __restrict__ A, const float* __restrict__ Bsw,
    const float* __restrict__ bias, float* __restrict__ C,
    int M, float ascale,
    const float* __restrict__ resid, const float* __restrict__ skip, int skipIdx)
{
    const int lane = threadIdx.x & 31;
    const int wave = threadIdx.x >> 5;
    const int task = blockIdx.x * 8 + wave;
    const int numTasks = (M >> 4) * 4;          // (M/16) m-tiles * 4 n-groups
    if (task >= numTasks) return;               // wave-uniform: EXEC all-1s
    const int mt = task >> 2;
    const int ng = task & 3;

    const int row = (mt << 4) + (lane & 15);
    const int kb  = (lane >> 4) << 1;           // 0 / 2

    const float* Arow = A + (size_t)row * DIM;
    const float* Bw   = Bsw + ((size_t)(ng << 6) * 32 + lane) * 8;

    v8f acc0 = {}, acc1 = {}, acc2 = {}, acc3 = {};
    for (int kc = 0; kc < 4; ++kc) {            // chunks of 16 k-steps (K=64)
        // prefetch next A chunk of this row (global_prefetch_b8)
        if (kc < 3) __builtin_prefetch(Arow + ((kc + 1) << 6), 0, 0);
#pragma unroll 4
        for (int ki = 0; ki < 16; ++ki) {
            const int ks = (kc << 4) + ki;
            v2f a = *(const v2f*)(Arow + (ks << 2) + kb);
            if (SCALE) { a.x *= ascale; a.y *= ascale; }
            const v4f blo = *(const v4f*)(Bw);      // b0 | b1
            const v4f bhi = *(const v4f*)(Bw + 4);  // b2 | b3
            Bw += 32 * 8;                           // next k-step, all lanes
            v2f b0; b0.x = blo.x; b0.y = blo.y;
            v2f b1; b1.x = blo.z; b1.y = blo.w;
            v2f b2; b2.x = bhi.x; b2.y = bhi.y;
            v2f b3; b3.x = bhi.z; b3.y = bhi.w;
            acc0 = __builtin_amdgcn_wmma_f32_16x16x4_f32(false, a, false, b0, (short)0, acc0, false, false);
            acc1 = __builtin_amdgcn_wmma_f32_16x16x4_f32(false, a, false, b1, (short)0, acc1, false, false);
            acc2 = __builtin_amdgcn_wmma_f32_16x16x4_f32(false, a, false, b2, (short)0, acc2, false, false);
            acc3 = __builtin_amdgcn_wmma_f32_16x16x4_f32(false, a, false, b3, (short)0, acc3, false, false);
        }
    }

    float alpha = 1.f, oneMinus = 0.f;
    if (SKIP) {
        const float g = skip[skipIdx];
        alpha    = 1.f / (1.f + __expf(-g));
        oneMinus = 1.f - alpha;
    }
    const int outRowBase = (mt << 4) + ((lane >> 4) << 3); // +8 for hi half
    const int cbase      = (ng << 6) + (lane & 15);
#pragma unroll
    for (int j = 0; j < 8; ++j) {
        const int r = outRowBase + j;
        float*       Cr = C + (size_t)r * DIM;
        const float* Hr = SKIP ? (resid + (size_t)r * DIM) : nullptr;
        float vals[4] = {acc0[j], acc1[j], acc2[j], acc3[j]};
#pragma unroll
        for (int t = 0; t < 4; ++t) {
            const int c = cbase + t * 16;
            float o = vals[t] + bias[c];
            if (SKIP) o = o * alpha + Hr[c] * oneMinus;
            Cr[c] = o;
        }
    }
}

// Order-preserving float -> uint key (memset(0) == -inf sentinel; any real
// score key > 0 so untouched segments are never consumed).
__device__ __forceinline__ unsigned f2key(float f) {
    unsigned b = __float_as_uint(f);
    return (b & 0x80000000u) ? ~b : (b | 0x80000000u);
}
__device__ __forceinline__ float key2f(unsigned k) {
    unsigned b = (k & 0x80000000u) ? (k & 0x7fffffffu) : ~k;
    return __uint_as_float(b);
}

// ---------------------------------------------------------------------------
// One wave per edge: s[e,h] = (q[dst] . k[src])_h * pri[h] / sqrt(dk),
// segment-max via keyed atomicMax. Lane = dk index (coalesced 128B lines).
// ---------------------------------------------------------------------------
__global__ __launch_bounds__(256) void edge_score_kernel(
    const float* __restrict__ q, const float* __restrict__ k,
    const int* __restrict__ src, const int* __restrict__ dst,
    const float* __restrict__ pri, float inv_sqrt_dk,
    float* __restrict__ sE, unsigned* __restrict__ smaxKey, int E)
{
    const int e    = (blockIdx.x * blockDim.x + threadIdx.x) >> 5;
    const int lane = threadIdx.x & 31;
    if (e >= E) return;
    const int s = src[e], d = dst[e];
    const float* qp = q + (size_t)d * DIM;
    const float* kp = k + (size_t)s * DIM;
#pragma unroll
    for (int h = 0; h < HEADS; ++h) {
        float p = qp[h * DK + lane] * kp[h * DK + lane];
        for (int off = 16; off; off >>= 1) p += __shfl_xor(p, off, 32);
        if (lane == 0) {
            float sc = p * pri[h] * inv_sqrt_dk;
            sE[(size_t)e * HEADS + h] = sc;
            atomicMax(&smaxKey[(size_t)d * HEADS + h], f2key(sc));
        }
    }
}

// ex = exp(s - smax[dst]); den[dst] += ex  (thread per edge-head)
__global__ __launch_bounds__(256) void edge_exp_kernel(
    const float* __restrict__ sE, const int* __restrict__ dst,
    const unsigned* __restrict__ smaxKey,
    float* __restrict__ exE, float* __restrict__ den, int E)
{
    int i = blockIdx.x * blockDim.x + threadIdx.x;
    if (i >= E * HEADS) return;
    int e = i >> 3, h = i & 7;
    int d = dst[e];
    float mx = key2f(smaxKey[(size_t)d * HEADS + h]);
    float ex = __expf(sE[i] - mx);
    exE[i] = ex;
    atomicAdd(&den[(size_t)d * HEADS + h], ex);
}

// t[dst] += v[src] * (ex / den[dst])  (one wave per edge; lane = dk index)
__global__ __launch_bounds__(256) void edge_agg_kernel(
    const float* __restrict__ v, const int* __restrict__ src,
    const int* __restrict__ dst, const float* __restrict__ exE,
    const float* __restrict__ den, float* __restrict__ t, int E)
{
    const int e    = (blockIdx.x * blockDim.x + threadIdx.x) >> 5;
    const int lane = threadIdx.x & 31;
    if (e >= E) return;
    const int s = src[e], d = dst[e];
    const float* vp = v + (size_t)s * DIM;
    float*       tp = t + (size_t)d * DIM;
#pragma unroll
    for (int h = 0; h < HEADS; ++h) {
        float a = exE[(size_t)e * HEADS + h] / den[(size_t)d * HEADS + h];
        atomicAdd(&tp[h * DK + lane], vp[h * DK + lane] * a);
    }
}

// ---------------------------------------------------------------------------
extern "C" void kernel_launch(void* const* d_in, const int* in_sizes, int n_in,
                              void* d_out, int out_size, void* d_ws, size_t ws_size,
                              hipStream_t stream)
{
    const float* h0      = (const float*)d_in[0];
    const float* h1      = (const float*)d_in[1];
    const float* Wk      = (const float*)d_in[2];
    const float* bk      = (const float*)d_in[3];
    const float* Wq      = (const float*)d_in[4];
    const float* bq      = (const float*)d_in[5];
    const float* Wv      = (const float*)d_in[6];
    const float* bv      = (const float*)d_in[7];
    const float* Wa      = (const float*)d_in[8];
    const float* ba      = (const float*)d_in[9];
    const float* rel_att = (const float*)d_in[10];
    const float* rel_msg = (const float*)d_in[11];
    const float* rel_pri = (const float*)d_in[12];
    const float* skip    = (const float*)d_in[13];
    const int*   srcs[3] = {(const int*)d_in[14], (const int*)d_in[16], (const int*)d_in[18]};
    const int*   dsts[3] = {(const int*)d_in[15], (const int*)d_in[17], (const int*)d_in[19]};
    float* out = (float*)d_out;

    const int N0 = in_sizes[0] / DIM;
    const int N1 = in_sizes[1] / DIM;
    const int E  = in_sizes[14];
    const int NMAX = (N0 > N1) ? N0 : N1;
    const size_t NF  = (size_t)NMAX * DIM;     // per-matrix floats (N0==N1)
    const size_t WSZ = (size_t)DIM * DIM;      // 65536

    // ---- workspace carve-up (floats) ----
    float* ws  = (float*)d_ws;
    float* q0  = ws;            // [N0,256]
    float* q1  = q0  + NF;
    float* ke0 = q1  + NF;      // k0 @ att0
    float* ke1 = ke0 + NF;      // k1 @ att1
    float* ke2 = ke1 + NF;      // k1 @ att2
    float* ve0 = ke2 + NF;      // v0 @ msg0
    float* ve1 = ve0 + NF;
    float* ve2 = ve1 + NF;
    float* t0  = ve2 + NF;      // agg target, node type 0 (relation 1)
    float* t1  = t0  + NF;      // agg target, node type 1 (relations 0+2)
    float* sE  = t1  + NF;                       // [E,8]
    float* exE = sE  + (size_t)E * HEADS;        // [E,8]
    unsigned* smaxKey = (unsigned*)(exE + (size_t)E * HEADS); // [NMAX,8]
    float* den = (float*)(smaxKey + (size_t)NMAX * HEADS);    // [NMAX,8]
    float* Wp  = den + (size_t)NMAX * HEADS;     // 6 x [256,256] folded weights
    float* bp  = Wp + 6 * WSZ;                   // 6 x [256]
    float* Bsw = bp + 6 * 256;                   // 10 x [256,256] swizzled
    // slots: 0=Wq0 1=Wq1 2..7=Wp0..5 8=Wa0 9=Wa1

    const dim3 blk(256);
    const float inv_sqrt_dk = 0.17677669529663687f; // 1/sqrt(32)

    // ---- 1) fold relation matrices into projection weights ----
    const int fgrid = (257 * 256 + 255) / 256;
    fuse_weight_kernel<<<fgrid, blk, 0, stream>>>(Wk,       bk,       rel_att,          Wp + 0*WSZ, bp + 0*256);
    fuse_weight_kernel<<<fgrid, blk, 0, stream>>>(Wk + WSZ, bk + 256, rel_att + 1*8192, Wp + 1*WSZ, bp + 1*256);
    fuse_weight_kernel<<<fgrid, blk, 0, stream>>>(Wk + WSZ, bk + 256, rel_att + 2*8192, Wp + 2*WSZ, bp + 2*256);
    fuse_weight_kernel<<<fgrid, blk, 0, stream>>>(Wv,       bv,       rel_msg,          Wp + 3*WSZ, bp + 3*256);
    fuse_weight_kernel<<<fgrid, blk, 0, stream>>>(Wv + WSZ, bv + 256, rel_msg + 1*8192, Wp + 4*WSZ, bp + 4*256);
    fuse_weight_kernel<<<fgrid, blk, 0, stream>>>(Wv + WSZ, bv + 256, rel_msg + 2*8192, Wp + 5*WSZ, bp + 5*256);

    // ---- 2) swizzle all GEMM weights into WMMA consumption order ----
    const int sgrid = (4 * 64 * 32 + 255) / 256;
    swizzle_B_kernel<<<sgrid, blk, 0, stream>>>(Wq,       Bsw + 0*WSZ);
    swizzle_B_kernel<<<sgrid, blk, 0, stream>>>(Wq + WSZ, Bsw + 1*WSZ);
    for (int i = 0; i < 6; ++i)
        swizzle_B_kernel<<<sgrid, blk, 0, stream>>>(Wp + i*WSZ, Bsw + (2 + i)*WSZ);
    swizzle_B_kernel<<<sgrid, blk, 0, stream>>>(Wa,       Bsw + 8*WSZ);
    swizzle_B_kernel<<<sgrid, blk, 0, stream>>>(Wa + WSZ, Bsw + 9*WSZ);

    // ---- 3) projections (WMMA GEMMs, no scale / no skip) ----
    auto gblocks = [](int M) { return ((M >> 4) * 4 + 7) / 8; };
    gemm256_wmma_kernel<false, false><<<gblocks(N0), blk, 0, stream>>>(h0, Bsw + 0*WSZ, bq,       q0,  N0, 1.f, nullptr, nullptr, 0);
    gemm256_wmma_kernel<false, false><<<gblocks(N1), blk, 0, stream>>>(h1, Bsw + 1*WSZ, bq + 256, q1,  N1, 1.f, nullptr, nullptr, 0);
    gemm256_wmma_kernel<false, false><<<gblocks(N0), blk, 0, stream>>>(h0, Bsw + 2*WSZ, bp + 0*256, ke0, N0, 1.f, nullptr, nullptr, 0);
    gemm256_wmma_kernel<false, false><<<gblocks(N1), blk, 0, stream>>>(h1, Bsw + 3*WSZ, bp + 1*256, ke1, N1, 1.f, nullptr, nullptr, 0);
    gemm256_wmma_kernel<false, false><<<gblocks(N1), blk, 0, stream>>>(h1, Bsw + 4*WSZ, bp + 2*256, ke2, N1, 1.f, nullptr, nullptr, 0);
    gemm256_wmma_kernel<false, false><<<gblocks(N0), blk, 0, stream>>>(h0, Bsw + 5*WSZ, bp + 3*256, ve0, N0, 1.f, nullptr, nullptr, 0);
    gemm256_wmma_kernel<false, false><<<gblocks(N1), blk, 0, stream>>>(h1, Bsw + 6*WSZ, bp + 4*256, ve1, N1, 1.f, nullptr, nullptr, 0);
    gemm256_wmma_kernel<false, false><<<gblocks(N1), blk, 0, stream>>>(h1, Bsw + 7*WSZ, bp + 5*256, ve2, N1, 1.f, nullptr, nullptr, 0);

    // ---- 4) zero aggregation targets ----
    hipMemsetAsync(t0, 0, 2 * NF * sizeof(float), stream);

    // ---- 5) edge softmax + aggregation per relation ----
    const float* qs[3]  = {q1, q0, q1};          // dst node type's q
    const float* kes[3] = {ke0, ke1, ke2};
    const float* ves[3] = {ve0, ve1, ve2};
    float*       ts[3]  = {t1, t0, t1};
    const int eWaveBlocks   = (E + 7) / 8;
    const int eThreadBlocks = (E * HEADS + 255) / 256;
    for (int r = 0; r < 3; ++r) {
        hipMemsetAsync(smaxKey, 0, (size_t)NMAX * HEADS * (sizeof(unsigned) + sizeof(float)), stream);
        edge_score_kernel<<<eWaveBlocks, blk, 0, stream>>>(
            qs[r], kes[r], srcs[r], dsts[r], rel_pri + r * HEADS, inv_sqrt_dk, sE, smaxKey, E);
        edge_exp_kernel<<<eThreadBlocks, blk, 0, stream>>>(sE, dsts[r], smaxKey, exE, den, E);
        edge_agg_kernel<<<eWaveBlocks, blk, 0, stream>>>(ves[r], srcs[r], dsts[r], exE, den, ts[r], E);
    }

    // ---- 6) output GEMMs with fused bias + sigmoid skip gate ----
    // out0 = (t0 @ Wa0 + ba0)*a0 + h0*(1-a0) ; t1 carries 0.5 relation-mean
    gemm256_wmma_kernel<false, true><<<gblocks(N0), blk, 0, stream>>>(
        t0, Bsw + 8*WSZ, ba,       out,                    N0, 1.0f, h0, skip, 0);
    gemm256_wmma_kernel<true, true><<<gblocks(N1), blk, 0, stream>>>(
        t1, Bsw + 9*WSZ, ba + 256, out + (size_t)N0 * DIM, N1, 0.5f, h1, skip, 1);
}